// GNN_LBA_57904749085258
// MI455X (gfx1250) — compile-verified
//
#include <hip/hip_runtime.h>
#include <cstddef>

typedef float v2f __attribute__((ext_vector_type(2)));
typedef float v8f __attribute__((ext_vector_type(8)));

#define BN_EPS 1e-5f

// ---------------- elementwise / setup kernels ----------------

__global__ void k_fill(float* __restrict__ p, float v, int n) {
  int i = blockIdx.x * blockDim.x + threadIdx.x;
  if (i < n) p[i] = v;
}

// deg pre-filled with 1.0 (self-loop weight); accumulate edge weights at col.
__global__ void k_deg(const int* __restrict__ col, const float* __restrict__ w,
                      float* __restrict__ deg, int E) {
  int e = blockIdx.x * blockDim.x + threadIdx.x;
  if (e < E) atomicAdd(&deg[col[e]], w[e]);
}

// deg -> deg^{-1/2} in place
__global__ void k_dis(float* __restrict__ deg, int N) {
  int i = blockIdx.x * blockDim.x + threadIdx.x;
  if (i < N) {
    float d = deg[i];
    deg[i] = (d > 0.f) ? rsqrtf(fmaxf(d, 1e-12f)) : 0.f;
  }
}

__global__ void k_norm(const int* __restrict__ row, const int* __restrict__ col,
                       const float* __restrict__ w, const float* __restrict__ dis,
                       float* __restrict__ norm, int E) {
  int e = blockIdx.x * blockDim.x + threadIdx.x;
  if (e < E) norm[e] = dis[row[e]] * w[e] * dis[col[e]];
}

// Pack W[K,Nc] -> Wp with adjacent K-rows interleaved:
// Wp[(kk*Nc + n)*2 + {0,1}] = W[2kk][n], W[2kk+1][n]
// so a WMMA B-fragment (two K-adjacent values for one column) is one b64 load.
__global__ void k_pack_b(const float* __restrict__ B, float* __restrict__ Bp,
                         int total2, int ncshift) {
  int i = blockIdx.x * blockDim.x + threadIdx.x;   // over (K/2)*Nc
  if (i < total2) {
    int kk = i >> ncshift;
    int n = i & ((1 << ncshift) - 1);
    int base = (kk << (ncshift + 1)) + n;          // 2kk * Nc + n
    v2f p;
    p.x = B[base];
    p.y = B[base + (1 << ncshift)];
    *(v2f*)(Bp + ((size_t)i << 1)) = p;
  }
}

// ---------------- fp32 WMMA GEMM: C[M,Nc] = A[M,K] @ Wp(packed B) --------
// one wave per 16x16 output tile; K stepped by 4 via V_WMMA_F32_16X16X4_F32.
// A row index is CLAMPED (not predicated): boundary tiles compute garbage in
// rows >= M which the guarded store never writes. Both A and B fragments are
// single b64 loads per step.
__global__ void k_gemm_wmma_f32(const float* __restrict__ A, const float* __restrict__ Bp,
                                float* __restrict__ C, int M, int K, int Nc) {
  const int lane = threadIdx.x & 31;
  const int wave = threadIdx.x >> 5;
  const int tilesN = Nc >> 4;
  const int tilesM = (M + 15) >> 4;
  const int tile = blockIdx.x * (blockDim.x >> 5) + wave;
  if (tile >= tilesM * tilesN) return;           // wave-uniform: EXEC stays all-ones
  const int tm = tile / tilesN;
  const int tn = tile - tm * tilesN;
  const int m0 = tm << 4, n0 = tn << 4;
  const int half = lane >> 4;                    // lanes 16-31 hold K+2,K+3 halves
  const int l = lane & 15;

  const int rowA = m0 + l;
  const int rowClamped = (rowA < M) ? rowA : (M - 1);
  const float* ap = A + (size_t)rowClamped * K + (half << 1);        // 8B-aligned
  const float* bp = Bp + (((size_t)half * Nc + n0 + l) << 1);        // packed pairs
  const int bstep = Nc << 2;                     // 2 pair-rows of Bp per K-step

  v8f acc = {0.f, 0.f, 0.f, 0.f, 0.f, 0.f, 0.f, 0.f};

  const int steps = K >> 2;                      // all K here are multiples of 16
#pragma unroll 4
  for (int s = 0; s < steps; ++s) {
    v2f a = *(const v2f*)ap;                     // A[row][k+2h .. k+2h+1]
    v2f b = *(const v2f*)bp;                     // B[k+2h][n], B[k+2h+1][n]
    acc = __builtin_amdgcn_wmma_f32_16x16x4_f32(
        false, a, false, b, (short)0, acc, false, false);
    ap += 4;
    bp += bstep;
  }

  // C/D layout: VGPR j holds row m0 + half*8 + j, col n0 + (lane&15)
#pragma unroll
  for (int j = 0; j < 8; ++j) {
    int r = m0 + (half << 3) + j;
    if (r < M) C[(size_t)r * Nc + n0 + l] = acc[j];
  }
}

// ---------------- aggregation (float4 vectorized) ----------------

// agg[i,:] = m[i,:] * dis[i]^2  (self-loop term; also initializes agg)
__global__ void k_self_agg4(const float4* __restrict__ m4, const float* __restrict__ dis,
                            float4* __restrict__ agg4, int total4, int qshift) {
  int i = blockIdx.x * blockDim.x + threadIdx.x;
  if (i < total4) {
    int n = i >> qshift;
    float d = dis[n];
    float s = d * d;
    float4 v = m4[i];
    v.x *= s; v.y *= s; v.z *= s; v.w *= s;
    agg4[i] = v;
  }
}

// agg[col[e],:] += m[row[e],:] * norm[e] ; one thread per (edge, float4 chunk)
__global__ void k_edge_agg4(const int* __restrict__ row, const int* __restrict__ col,
                            const float* __restrict__ norm, const float4* __restrict__ m4,
                            float* __restrict__ agg, int total4, int qshift) {
  int i = blockIdx.x * blockDim.x + threadIdx.x;
  if (i < total4) {
    int e = i >> qshift;
    int c = i & ((1 << qshift) - 1);
    int q = 1 << qshift;
    float w = norm[e];
    float4 v = m4[(size_t)row[e] * q + c];
    float* dst = agg + (((size_t)col[e] * q + c) << 2);
    atomicAdd(dst + 0, v.x * w);
    atomicAdd(dst + 1, v.y * w);
    atomicAdd(dst + 2, v.z * w);
    atomicAdd(dst + 3, v.w * w);
  }
}

// ---------------- batch norm (training-mode batch stats) ----------------

__global__ void k_bn_stats(const float* __restrict__ agg, const float* __restrict__ bias,
                           float* __restrict__ csum, float* __restrict__ csumsq,
                           int N, int fshift, int preRelu) {
  int g = blockIdx.x * blockDim.x + threadIdx.x;   // T = dout * 512 threads
  int f = g & ((1 << fshift) - 1);
  int r = g >> fshift;
  float bf = bias[f];
  float s = 0.f, s2 = 0.f;
  for (; r < N; r += 512) {
    float t = agg[((size_t)r << fshift) + f] + bf;
    if (preRelu) t = fmaxf(t, 0.f);
    s += t;
    s2 += t * t;
  }
  atomicAdd(&csum[f], s);
  atomicAdd(&csumsq[f], s2);
}

__global__ void k_bn_apply(const float* __restrict__ agg, const float* __restrict__ bias,
                           const float* __restrict__ csum, const float* __restrict__ csumsq,
                           const float* __restrict__ gam, const float* __restrict__ bet,
                           float* __restrict__ out, int N, int total, int fshift,
                           int preRelu, int postRelu) {
  int i = blockIdx.x * blockDim.x + threadIdx.x;
  if (i < total) {
    int f = i & ((1 << fshift) - 1);
    float t = agg[i] + bias[f];
    if (preRelu) t = fmaxf(t, 0.f);
    float invN = 1.f / (float)N;
    float mu = csum[f] * invN;
    float var = csumsq[f] * invN - mu * mu;
    float y = gam[f] * (t - mu) * rsqrtf(var + BN_EPS) + bet[f];
    if (postRelu) y = fmaxf(y, 0.f);
    out[i] = y;
  }
}

// ---------------- pooling + MLP head ----------------

__global__ void k_pool4(const float4* __restrict__ h4, const int* __restrict__ batch,
                        float* __restrict__ pooled, int total4) {
  int i = blockIdx.x * blockDim.x + threadIdx.x;
  if (i < total4) {
    int n = i >> 5;
    int c = i & 31;
    float4 v = h4[i];
    float* dst = pooled + (((size_t)batch[n] << 5) + c) * 4;
    atomicAdd(dst + 0, v.x);
    atomicAdd(dst + 1, v.y);
    atomicAdd(dst + 2, v.z);
    atomicAdd(dst + 3, v.w);
  }
}

__global__ void k_relu_inplace(float* __restrict__ p, int n) {
  int i = blockIdx.x * blockDim.x + threadIdx.x;
  if (i < n) p[i] = fmaxf(p[i], 0.f);
}

__global__ void k_bias_relu(const float* __restrict__ in, const float* __restrict__ bias,
                            float* __restrict__ out, int n, int fshift) {
  int i = blockIdx.x * blockDim.x + threadIdx.x;
  if (i < n) out[i] = fmaxf(in[i] + bias[i & ((1 << fshift) - 1)], 0.f);
}

__global__ void k_fc2(const float* __restrict__ p1, const float* __restrict__ w,
                      const float* __restrict__ b, float* __restrict__ out, int G, int K) {
  int g = blockIdx.x * blockDim.x + threadIdx.x;
  if (g < G) {
    float s = b[0];
    for (int k = 0; k < K; ++k) s += p1[(size_t)g * K + k] * w[k];
    out[g] = s;
  }
}

// ---------------- host side ----------------

static inline int ceil_div(int a, int b) { return (a + b - 1) / b; }
static inline int ilog2(int v) { int s = 0; while ((1 << s) < v) ++s; return s; }

extern "C" void kernel_launch(void* const* d_in, const int* in_sizes, int n_in,
                              void* d_out, int out_size, void* d_ws, size_t ws_size,
                              hipStream_t stream) {
  const float* x     = (const float*)d_in[0];
  const int*   ei    = (const int*)d_in[1];
  const float* ew    = (const float*)d_in[2];
  const int*   batch = (const int*)d_in[3];
  const float* W[5], *bb[5], *gg[5], *bt[5];
  for (int l = 0; l < 5; ++l) {
    W[l]  = (const float*)d_in[4 + 4 * l];
    bb[l] = (const float*)d_in[5 + 4 * l];
    gg[l] = (const float*)d_in[6 + 4 * l];
    bt[l] = (const float*)d_in[7 + 4 * l];
  }
  const float* fc1w = (const float*)d_in[24];
  const float* fc1b = (const float*)d_in[25];
  const float* fc2w = (const float*)d_in[26];
  const float* fc2b = (const float*)d_in[27];
  float* out = (float*)d_out;

  const int FIN = 128;
  const int N = in_sizes[0] / FIN;
  const int E = in_sizes[2];
  const int G = 256;
  static const int dims[6] = {128, 16, 32, 64, 64, 128};

  // workspace carve-up (floats)
  float* ws = (float*)d_ws;
  float* deg    = ws;                            // N   (becomes dis in place)
  float* norm   = deg + N;                       // E
  float* bufM   = norm + E;                      // N*128  (GEMM output m)
  float* bufAgg = bufM + (size_t)N * 128;        // N*128  (aggregation)
  float* bufH   = bufAgg + (size_t)N * 128;      // N*128  (layer activations)
  float* colS   = bufH + (size_t)N * 128;        // 128
  float* colQ   = colS + 128;                    // 128
  float* pooled = colQ + 128;                    // G*128
  float* fcm    = pooled + (size_t)G * 128;      // G*64
  float* p1     = fcm + (size_t)G * 64;          // G*64
  float* Wp     = p1 + (size_t)G * 64;           // 128*128 (packed weights)

  const int T = 256;

  // ---- gcn_norm
  const int* row = ei;
  const int* col = ei + E;
  k_fill<<<ceil_div(N, T), T, 0, stream>>>(deg, 1.0f, N);
  k_deg<<<ceil_div(E, T), T, 0, stream>>>(col, ew, deg, E);
  k_dis<<<ceil_div(N, T), T, 0, stream>>>(deg, N);           // deg -> dis in place
  k_norm<<<ceil_div(E, T), T, 0, stream>>>(row, col, ew, deg, norm, E);

  // ---- 5 GCN layers
  const float* hin = x;
  for (int l = 0; l < 5; ++l) {
    const int din = dims[l], dout = dims[l + 1];
    const int fshift = ilog2(dout);
    const int qshift = fshift - 2;               // float4 chunks per row

    // m = h @ W  (fp32 WMMA, packed-B)
    {
      int total2 = (din >> 1) * dout;
      k_pack_b<<<ceil_div(total2, T), T, 0, stream>>>(W[l], Wp, total2, fshift);
      int tiles = ceil_div(N, 16) * (dout / 16);
      k_gemm_wmma_f32<<<ceil_div(tiles, T / 32), T, 0, stream>>>(
          hin, Wp, bufM, N, din, dout);
    }
    // agg = selfloop init + scatter-add over edges (float4 path)
    {
      int totalN4 = N * (dout >> 2);
      int totalE4 = E * (dout >> 2);
      k_self_agg4<<<ceil_div(totalN4, T), T, 0, stream>>>(
          (const float4*)bufM, deg, (float4*)bufAgg, totalN4, qshift);
      k_edge_agg4<<<ceil_div(totalE4, T), T, 0, stream>>>(
          row, col, norm, (const float4*)bufM, bufAgg, totalE4, qshift);
    }

    // batch-norm (+bias, relu ordering per layer)
    const int preRelu  = (l < 3) ? 1 : 0;        // layers 1-3: relu before bn
    const int postRelu = (l == 3) ? 1 : 0;       // layer 4: relu after bn
    k_fill<<<1, 256, 0, stream>>>(colS, 0.f, 256);   // colS+colQ contiguous
    k_bn_stats<<<dout * 2, 256, 0, stream>>>(bufAgg, bb[l], colS, colQ, N, fshift, preRelu);
    k_bn_apply<<<ceil_div(N * dout, T), T, 0, stream>>>(
        bufAgg, bb[l], colS, colQ, gg[l], bt[l], bufH, N, N * dout, fshift, preRelu, postRelu);

    hin = bufH;
  }

  // ---- global_add_pool -> relu -> fc1(WMMA) -> relu -> fc2
  k_fill<<<ceil_div(G * 128, T), T, 0, stream>>>(pooled, 0.f, G * 128);
  k_pool4<<<ceil_div(N * 32, T), T, 0, stream>>>((const float4*)bufH, batch, pooled, N * 32);
  k_relu_inplace<<<ceil_div(G * 128, T), T, 0, stream>>>(pooled, G * 128);

  {
    int total2 = (128 >> 1) * 64;                // pack fc1_w (128x64)
    k_pack_b<<<ceil_div(total2, T), T, 0, stream>>>(fc1w, Wp, total2, 6);
    int tiles = (G / 16) * (64 / 16);            // 256x128 @ 128x64
    k_gemm_wmma_f32<<<ceil_div(tiles, T / 32), T, 0, stream>>>(pooled, Wp, fcm, G, 128, 64);
  }
  k_bias_relu<<<ceil_div(G * 64, T), T, 0, stream>>>(fcm, fc1b, p1, G * 64, 6);
  k_fc2<<<1, G, 0, stream>>>(p1, fc2w, fc2b, out, G, 64);
}